// erf_loss_18253611008030
// MI455X (gfx1250) — compile-verified
//
#include <hip/hip_runtime.h>
#include <hip/hip_bf16.h>
#include <math.h>

// Problem constants from the reference.
#define DIMD   512
#define CROWS  85742
#define BROWS  256
#define NELEM  (BROWS * DIMD)      // 131072
#define NCHUNK 96                  // row chunks for colsum
#define ROWS_PER_CHUNK ((CROWS + NCHUNK - 1) / NCHUNK)  // 894
#define NB3    128                 // blocks for dist / loss partials
#define UNROLL 8                   // row-groups (of 4 rows) per macro-iter

typedef __attribute__((ext_vector_type(2))) float v2f;
typedef __attribute__((ext_vector_type(8))) float v8f;

__device__ __forceinline__ v8f wmma_acc(v2f a, v2f b, v8f c) {
    return __builtin_amdgcn_wmma_f32_16x16x4_f32(
        /*neg_a=*/false, a, /*neg_b=*/false, b,
        /*c_mod=*/(short)0, c, /*reuse_a=*/false, /*reuse_b=*/false);
}

// ---------------------------------------------------------------------------
// Kernel 1: column sum of weight[C, D] via V_WMMA_F32_16X16X4_F32.
//   ones(16x4) @ Wtile(4x16) accumulated into C — every output row M equals
//   the column sum, so we read row M=0 (lanes 0-15, VGPR0 of the v8f acc).
// grid = (32 column tiles, NCHUNK row chunks), block = 256 (8 wave32s).
// Each wave owns row-groups rowStart + wid*4 + 32*k. The bulk loop issues
// 16 independent global_load_b32 per lane (64 B in flight per lane) before
// the 8 WMMAs, maximizing MLP; the tail loop carries the range guards.
// ---------------------------------------------------------------------------
__global__ void colsum_wmma(const float* __restrict__ w,
                            float* __restrict__ partials) {
    const int ct   = blockIdx.x;          // column tile: 16 columns
    const int rc   = blockIdx.y;          // row chunk
    const int tid  = threadIdx.x;
    const int wid  = tid >> 5;            // wave id 0..7
    const int lane = tid & 31;
    const int half = lane >> 4;           // 0: K=0/1 rows, 1: K=2/3 rows
    const int n    = lane & 15;           // column within tile
    const int col  = ct * 16 + n;
    const int rowOff = half * 2;

    const int rowStart = rc * ROWS_PER_CHUNK;
    int rowEnd = rowStart + ROWS_PER_CHUNK;
    if (rowEnd > CROWS) rowEnd = CROWS;

    v2f a  = {1.0f, 1.0f};                // ones A-matrix (16x4, all lanes 1)
    v8f c0 = {0.f, 0.f, 0.f, 0.f, 0.f, 0.f, 0.f, 0.f};
    v8f c1 = {0.f, 0.f, 0.f, 0.f, 0.f, 0.f, 0.f, 0.f};

    int r = rowStart + wid * 4;

    // Bulk: UNROLL full groups, branch-free loads, 2 accumulators for ILP.
    for (; r + (UNROLL - 1) * 32 + 4 <= rowEnd; r += UNROLL * 32) {
        v2f b[UNROLL];
#pragma unroll
        for (int u = 0; u < UNROLL; ++u) {
            const int row0 = r + u * 32 + rowOff;
            b[u].x = w[row0 * DIMD + col];           // K = half*2
            b[u].y = w[(row0 + 1) * DIMD + col];     // K = half*2 + 1
        }
#pragma unroll
        for (int u = 0; u < UNROLL; ++u) {
            if (u & 1) c1 = wmma_acc(a, b[u], c1);
            else       c0 = wmma_acc(a, b[u], c0);
        }
    }

    // Tail: remaining (<UNROLL) groups, rows guarded against chunk end.
    for (; r < rowEnd; r += 32) {
        const int row0 = r + rowOff;
        const int row1 = row0 + 1;
        v2f b;
        b.x = (row0 < rowEnd) ? w[row0 * DIMD + col] : 0.0f;
        b.y = (row1 < rowEnd) ? w[row1 * DIMD + col] : 0.0f;
        c0 = wmma_acc(a, b, c0);
    }

    // Row M=0 of each accumulator lives in element 0 of lanes 0..15.
    const float mysum = c0[0] + c1[0];
    __shared__ float acc[8][16];
    if (lane < 16) acc[wid][n] = mysum;
    __syncthreads();

    if (tid < 16) {
        float s = 0.0f;
#pragma unroll
        for (int wv = 0; wv < 8; ++wv) s += acc[wv][tid];
        partials[rc * DIMD + ct * 16 + tid] = s;
    }
}

// ---------------------------------------------------------------------------
// Kernel 2: S[d] = sum over NCHUNK chunk-partials. 2 blocks x 256 threads.
// ---------------------------------------------------------------------------
__global__ void reduce_colsum(const float* __restrict__ partials,
                              float* __restrict__ S) {
    const int d = blockIdx.x * blockDim.x + threadIdx.x;
    if (d < DIMD) {
        float s = 0.0f;
        for (int i = 0; i < NCHUNK; ++i) s += partials[i * DIMD + d];
        S[d] = s;
    }
}

// ---------------------------------------------------------------------------
// Kernel 3: block partials of sum((wy-mu)^2) and sum((wi-mu)^2),
// wy = weight[label[b], d], wi = S[d] - wy.  NB3 blocks x 256 threads.
// ---------------------------------------------------------------------------
__global__ void dist_partials(const float* __restrict__ w,
                              const float* __restrict__ mu,
                              const int* __restrict__ label,
                              const float* __restrict__ S,
                              float* __restrict__ dyPart,
                              float* __restrict__ diPart) {
    const int tid = threadIdx.x;
    float sy = 0.0f, si = 0.0f;
    for (int i = blockIdx.x * blockDim.x + tid; i < NELEM;
         i += gridDim.x * blockDim.x) {
        const int b = i >> 9;         // / 512
        const int d = i & (DIMD - 1); // % 512
        const float wy = w[(size_t)label[b] * DIMD + d];
        const float m  = mu[i];
        const float dy = wy - m;
        const float di = (S[d] - wy) - m;
        sy += dy * dy;
        si += di * di;
    }
    __shared__ float saccY[256];
    __shared__ float saccI[256];
    saccY[tid] = sy;
    saccI[tid] = si;
    __syncthreads();
    for (int off = 128; off > 0; off >>= 1) {
        if (tid < off) {
            saccY[tid] += saccY[tid + off];
            saccI[tid] += saccI[tid + off];
        }
        __syncthreads();
    }
    if (tid == 0) {
        dyPart[blockIdx.x] = saccY[0];
        diPart[blockIdx.x] = saccI[0];
    }
}

// ---------------------------------------------------------------------------
// Kernel 4: every block redundantly reduces the NB3 dist partials (cheap,
// deterministic), then accumulates erf(distY/(sqrt2*std)) +
// erfc(distI/(sqrt2*std)) over its grid-stride slice.  NB3 blocks x 256.
// ---------------------------------------------------------------------------
__global__ void loss_partials(const float* __restrict__ stdv,
                              const float* __restrict__ dyPart,
                              const float* __restrict__ diPart,
                              float* __restrict__ lossPart) {
    const int tid = threadIdx.x;
    __shared__ float sdy[NB3];
    __shared__ float sdi[NB3];
    if (tid < NB3) {
        sdy[tid] = dyPart[tid];
        sdi[tid] = diPart[tid];
    }
    __syncthreads();
    for (int off = NB3 / 2; off > 0; off >>= 1) {
        if (tid < off) {
            sdy[tid] += sdy[tid + off];
            sdi[tid] += sdi[tid + off];
        }
        __syncthreads();
    }
    const float inv_sqrt2 = 0.70710678118654752440f;
    const float ay = sqrtf(sdy[0]) * inv_sqrt2;  // dist_y / sqrt(2)
    const float ai = sqrtf(sdi[0]) * inv_sqrt2;  // dist_i / sqrt(2)

    float s = 0.0f;
    for (int i = blockIdx.x * blockDim.x + tid; i < NELEM;
         i += gridDim.x * blockDim.x) {
        const float inv = 1.0f / stdv[i];        // uniform(0,1): positive
        s += erff(ay * inv) + erfcf(ai * inv);
    }
    __shared__ float sacc[256];
    sacc[tid] = s;
    __syncthreads();
    for (int off = 128; off > 0; off >>= 1) {
        if (tid < off) sacc[tid] += sacc[tid + off];
        __syncthreads();
    }
    if (tid == 0) lossPart[blockIdx.x] = sacc[0];
}

// ---------------------------------------------------------------------------
// Kernel 5: final NB3 -> 1 reduction into d_out[0].
// ---------------------------------------------------------------------------
__global__ void final_reduce(const float* __restrict__ lossPart,
                             float* __restrict__ out) {
    const int tid = threadIdx.x;   // 128 threads
    __shared__ float sacc[NB3];
    sacc[tid] = lossPart[tid];
    __syncthreads();
    for (int off = NB3 / 2; off > 0; off >>= 1) {
        if (tid < off) sacc[tid] += sacc[tid + off];
        __syncthreads();
    }
    if (tid == 0) out[0] = sacc[0];
}

// ---------------------------------------------------------------------------
// Launch: inputs are {x, mu, std, weight, label}; x is unused by the math.
// ---------------------------------------------------------------------------
extern "C" void kernel_launch(void* const* d_in, const int* in_sizes, int n_in,
                              void* d_out, int out_size, void* d_ws, size_t ws_size,
                              hipStream_t stream) {
    (void)in_sizes; (void)n_in; (void)out_size; (void)ws_size;

    const float* mu     = (const float*)d_in[1];
    const float* stdv   = (const float*)d_in[2];
    const float* weight = (const float*)d_in[3];
    const int*   label  = (const int*)d_in[4];
    float*       out    = (float*)d_out;

    // Workspace layout (floats): chunk partials | S | dyPart | diPart | lossPart
    float* wsf      = (float*)d_ws;
    float* partials = wsf;                         // NCHUNK * 512
    float* S        = partials + NCHUNK * DIMD;    // 512
    float* dyPart   = S + DIMD;                    // NB3
    float* diPart   = dyPart + NB3;                // NB3
    float* lossPart = diPart + NB3;                // NB3

    colsum_wmma<<<dim3(DIMD / 16, NCHUNK), 256, 0, stream>>>(weight, partials);
    reduce_colsum<<<2, 256, 0, stream>>>(partials, S);
    dist_partials<<<NB3, 256, 0, stream>>>(weight, mu, label, S, dyPart, diPart);
    loss_partials<<<NB3, 256, 0, stream>>>(stdv, dyPart, diPart, lossPart);
    final_reduce<<<1, NB3, 0, stream>>>(lossPart, out);
}